// RepulsionLoss_66563403153930
// MI455X (gfx1250) — compile-verified
//
#include <hip/hip_runtime.h>

typedef float v2f __attribute__((ext_vector_type(2)));
typedef float v8f __attribute__((ext_vector_type(8)));

#define BATCH 4
#define NPTS  4096
#define KNN   10
#define HRAD  0.0005f
#define QPW   16            // queries per wave (one 16-wide N tile)
#define WAVES 8             // waves per workgroup (256 threads, wave32)
#define QPB   (QPW * WAVES) // 128 queries per block
#define TILES (NPTS / 16)   // 256 candidate tiles

__global__ void RepulsionLoss_zero_kernel(float* out) { out[0] = 0.0f; }

__global__ __launch_bounds__(256)
void RepulsionLoss_66563403153930_kernel(const float* __restrict__ x,
                                         float* __restrict__ out) {
    // 64 KB LDS. Front 48 KB is first used as the raw xyz staging area
    // (filled by async global->LDS DMA), then repacked in place to
    // (x, y, z, |p|^2) float4 records.
    __shared__ __align__(16) float s_mem[NPTS * 4];
    float4* s_pts = (float4*)s_mem;

    const int tid = threadIdx.x;
    const int b   = blockIdx.x / (NPTS / QPB);
    const int qt  = blockIdx.x % (NPTS / QPB);
    const float* xb = x + (size_t)b * NPTS * 3;

    // ---- stage 1a: async global -> LDS copy of the 48 KB coordinate slab ----
    for (int c = tid; c < (NPTS * 3) / 4; c += 256) {
        unsigned           ldsa = (unsigned)(size_t)&s_mem[c * 4];
        unsigned long long ga   = (unsigned long long)(size_t)(xb + c * 4);
        asm volatile("global_load_async_to_lds_b128 %0, %1, off"
                     :: "v"(ldsa), "v"(ga) : "memory");
    }
    asm volatile("s_wait_asynccnt 0x0" ::: "memory");
    __syncthreads();

    // ---- stage 1b: in-place repack, descending rounds ----
    // Packed float4 write of point p clobbers raw triplets of indices
    // >= 4p/3 > p only, i.e. points consumed in earlier (higher) rounds.
    for (int k = (NPTS / 256) - 1; k >= 0; --k) {
        const int p = k * 256 + tid;
        float px = s_mem[p * 3 + 0];
        float py = s_mem[p * 3 + 1];
        float pz = s_mem[p * 3 + 2];
        __syncthreads();
        s_pts[p] = make_float4(px, py, pz, px * px + py * py + pz * pz);
        __syncthreads();
    }

    // ---- stage 2: WMMA distance tiles + register-resident top-K ----
    const int lane = tid & 31;
    const int wave = tid >> 5;
    const int l15  = lane & 15;
    const int half = lane >> 4;

    const int q = qt * QPB + wave * QPW + l15;
    const float4 qp = s_pts[q];

    // B (4x16, K x N): lane = N, halves hold K{0,1}/{2,3}.
    // B[0..2][n] = q_n, B[3][n] = 1 (picks up |c|^2 from A's K=3 slot).
    v2f bmat;
    bmat[0] = half ? qp.z : qp.x;
    bmat[1] = half ? 1.0f : qp.y;

    v8f cinit; // C = |q|^2 splat  ->  D = |c|^2 + |q|^2 - 2 c.q
#pragma unroll
    for (int j = 0; j < 8; ++j) cinit[j] = qp.w;

    float best[KNN]; // sorted ascending; best[9] = current 10th smallest
#pragma unroll
    for (int i = 0; i < KNN; ++i) best[i] = __builtin_inff();

    auto scan8 = [&](v8f d) {
        // one branch per 8 distances: min-tree, then rare insertion path
        float m = fminf(fminf(fminf(d[0], d[1]), fminf(d[2], d[3])),
                        fminf(fminf(d[4], d[5]), fminf(d[6], d[7])));
        if (fmaxf(m, 0.0f) < best[KNN - 1]) {
#pragma unroll
            for (int j = 0; j < 8; ++j) {
                float v = fmaxf(d[j], 0.0f);
                if (v < best[KNN - 1]) {
#pragma unroll
                    for (int i = 0; i < KNN; ++i) {
                        float lo = fminf(best[i], v);
                        v        = fmaxf(best[i], v);
                        best[i]  = lo;
                    }
                }
            }
        }
    };

    for (int t = 0; t < TILES; t += 2) {
        const float4 cp0 = s_pts[t * 16 + l15];
        const float4 cp1 = s_pts[t * 16 + 16 + l15];
        // A (16x4, M x K): lane(&15) = M, halves hold K{0,1}/{2,3}.
        // A[m][0..2] = -2*c_m, A[m][3] = |c_m|^2
        v2f a0, a1;
        a0[0] = half ? (-2.0f * cp0.z) : (-2.0f * cp0.x);
        a0[1] = half ? cp0.w           : (-2.0f * cp0.y);
        a1[0] = half ? (-2.0f * cp1.z) : (-2.0f * cp1.x);
        a1[1] = half ? cp1.w           : (-2.0f * cp1.y);

        v8f d0 = __builtin_amdgcn_wmma_f32_16x16x4_f32(
            false, a0, false, bmat, (short)0, cinit, false, false);
        v8f d1 = __builtin_amdgcn_wmma_f32_16x16x4_f32(
            false, a1, false, bmat, (short)0, cinit, false, false);

        scan8(d0);
        scan8(d1);
    }

    // ---- stage 3: merge the two half-lane lists (bitonic halver) ----
    float other[KNN];
#pragma unroll
    for (int i = 0; i < KNN; ++i) other[i] = __shfl_xor(best[i], 16, 32);

    float ten[KNN]; // 10 smallest of the union of two sorted 10-lists
#pragma unroll
    for (int i = 0; i < KNN; ++i) ten[i] = fminf(best[i], other[KNN - 1 - i]);

    float mn = ten[0];
#pragma unroll
    for (int i = 1; i < KNN; ++i) mn = fminf(mn, ten[i]);

    float s = 0.0f;
#pragma unroll
    for (int i = 0; i < KNN; ++i) s += fmaxf(HRAD - ten[i], 0.0f);
    s -= fmaxf(HRAD - mn, 0.0f);  // drop the self neighbor (global minimum)

    if (half) s = 0.0f;           // upper half-lane duplicates the result
    s *= (1.0f / (float)(BATCH * NPTS * (KNN - 1)));

#pragma unroll
    for (int off = 16; off >= 1; off >>= 1) s += __shfl_xor(s, off, 32);

    if (lane == 0) atomicAdd(out, s);
}

extern "C" void kernel_launch(void* const* d_in, const int* in_sizes, int n_in,
                              void* d_out, int out_size, void* d_ws, size_t ws_size,
                              hipStream_t stream) {
    (void)in_sizes; (void)n_in; (void)out_size; (void)d_ws; (void)ws_size;
    const float* xin = (const float*)d_in[0];
    float* out = (float*)d_out;

    RepulsionLoss_zero_kernel<<<1, 1, 0, stream>>>(out);
    RepulsionLoss_66563403153930_kernel<<<BATCH * (NPTS / QPB), 256, 0, stream>>>(xin, out);
}